// GraphAttentionLayer_57397942944040
// MI455X (gfx1250) — compile-verified
//
#include <hip/hip_runtime.h>
#include <hip/hip_bf16.h>

// ---------------------------------------------------------------------------
// Stage 1: complex GEMV  tmp = [Wh_r | Wh_i] @ [[Wr, Wi], [-Wi, Wr]] + b
// via V_WMMA_F32_16X16X4_F32 (fp32 WMMA, wave32).
//   - one wave computes 16 nodes
//   - A tile: 16 nodes x 4 consecutive channels (lane l: row = l&15,
//     K pair = 2*(l>>4))   [ISA 7.12.2, 32-bit A 16x4 layout]
//   - B tile: 4x16, only columns 0,1 non-zero (weight chunk / complex combo)
//   - two WMMAs per k-step share one accumulator:
//       acc += A_r x B_r   (B_r cols:  W_r,  W_i)
//       acc += A_i x B_i   (B_i cols: -W_i,  W_r)
//     -> acc col 0 = tmp_r, col 1 = tmp_i for the 16 nodes
// ---------------------------------------------------------------------------

typedef __attribute__((ext_vector_type(2))) float v2f;
typedef __attribute__((ext_vector_type(8))) float v8f;

#define MAX_C 2048  // LDS staging capacity for W_real/W_imag (C=256 in ref)

__global__ void gemv_complex_wmma(const float* __restrict__ Whr,
                                  const float* __restrict__ Whi,
                                  const float* __restrict__ wr,
                                  const float* __restrict__ wi,
                                  const float* __restrict__ br,
                                  const float* __restrict__ bi,
                                  float* __restrict__ tmp_r,
                                  float* __restrict__ tmp_i,
                                  int N, int C) {
  __shared__ float s_wr[MAX_C];
  __shared__ float s_wi[MAX_C];
  for (int i = threadIdx.x; i < C; i += blockDim.x) {
    s_wr[i] = wr[i];
    s_wi[i] = wi[i];
  }
  __syncthreads();

  const int lane = threadIdx.x & 31;
  const int wave = threadIdx.x >> 5;
  const int m    = lane & 15;  // A row / B-D column index within tile
  const int kh   = lane >> 4;  // which K-half this lane carries
  const int nodeBase = (blockIdx.x * (int)(blockDim.x >> 5) + wave) * 16;
  if (nodeBase >= N) return;   // wave-uniform exit: EXEC stays all-ones

  int row = nodeBase + m;
  if (row >= N) row = N - 1;   // clamp loads; stores are guarded below
  const float* pr = Whr + (size_t)row * C + 2 * kh;
  const float* pi = Whi + (size_t)row * C + 2 * kh;

  v8f acc = {0.f, 0.f, 0.f, 0.f, 0.f, 0.f, 0.f, 0.f};

  for (int k0 = 0; k0 < C; k0 += 4) {
    v2f a_r = *(const v2f*)(pr + k0);   // A row m, K = k0+2kh .. k0+2kh+1
    v2f a_i = *(const v2f*)(pi + k0);

    const int kk  = k0 + 2 * kh;        // B row index carried by this lane
    const float wr0 = s_wr[kk], wr1 = s_wr[kk + 1];
    const float wi0 = s_wi[kk], wi1 = s_wi[kk + 1];

    v2f Br, Bi;                          // B[2kh][m], B[2kh+1][m]
    Br.x = (m == 0) ?  wr0 : ((m == 1) ? wi0 : 0.f);
    Br.y = (m == 0) ?  wr1 : ((m == 1) ? wi1 : 0.f);
    Bi.x = (m == 0) ? -wi0 : ((m == 1) ? wr0 : 0.f);
    Bi.y = (m == 0) ? -wi1 : ((m == 1) ? wr1 : 0.f);

    acc = __builtin_amdgcn_wmma_f32_16x16x4_f32(
        /*neg_a=*/false, a_r, /*neg_b=*/false, Br,
        /*c_mod=*/(short)0, acc, /*reuse_a=*/false, /*reuse_b=*/false);
    acc = __builtin_amdgcn_wmma_f32_16x16x4_f32(
        false, a_i, false, Bi, (short)0, acc, false, false);
  }

  // D layout (16x16 f32): lane l, VGPR v -> row (v + 8*(l>>4)), col (l&15).
  // Column 0 = tmp_r, column 1 = tmp_i  -> lanes 0,1,16,17 hold the results.
  if (m < 2) {
    const float bias = (m == 0) ? br[0] : bi[0];
    float* dst = (m == 0) ? tmp_r : tmp_i;
#pragma unroll
    for (int v = 0; v < 8; v++) {
      const int node = nodeBase + v + 8 * kh;
      if (node < N) dst[node] = acc[v] + bias;
    }
  }
}

// ---------------------------------------------------------------------------
// Stage 2: attention + column normalization.
// One thread per column m. tmp_* is 800 KB -> L2 resident, gathers are cheap.
// K=32 path keeps all attention values in registers (fully unrolled).
// ---------------------------------------------------------------------------

template <int KC>
__global__ void attention_fixed(const int* __restrict__ Nn,
                                const float* __restrict__ tmp_r,
                                const float* __restrict__ tmp_i,
                                float* __restrict__ out, int M) {
  const int m0 = blockIdx.x * blockDim.x + threadIdx.x;
  if (m0 >= M) return;
  const int c = Nn[m0];
  const float ar = tmp_r[c];
  const float ai = tmp_i[c];
  float att[KC];
  float sum = 0.f;
#pragma unroll
  for (int k = 0; k < KC; k++) {
    const int n = Nn[(size_t)(k + 1) * M + m0];
    const float a = fmaxf(fmaf(ar, tmp_r[n], ai * tmp_i[n]), 0.f);
    att[k] = a;
    sum += a;
  }
  const float inv = 1.0f / (sum + 0.001f);
#pragma unroll
  for (int k = 0; k < KC; k++) out[(size_t)k * M + m0] = att[k] * inv;
}

__global__ void attention_generic(const int* __restrict__ Nn,
                                  const float* __restrict__ tmp_r,
                                  const float* __restrict__ tmp_i,
                                  float* __restrict__ out, int M, int K) {
  const int m0 = blockIdx.x * blockDim.x + threadIdx.x;
  if (m0 >= M) return;
  const int c = Nn[m0];
  const float ar = tmp_r[c];
  const float ai = tmp_i[c];
  float sum = 0.f;
  for (int k = 0; k < K; k++) {
    const int n = Nn[(size_t)(k + 1) * M + m0];
    sum += fmaxf(fmaf(ar, tmp_r[n], ai * tmp_i[n]), 0.f);
  }
  const float inv = 1.0f / (sum + 0.001f);
  for (int k = 0; k < K; k++) {
    const int n = Nn[(size_t)(k + 1) * M + m0];
    const float a = fmaxf(fmaf(ar, tmp_r[n], ai * tmp_i[n]), 0.f);
    out[(size_t)k * M + m0] = a * inv;
  }
}

// ---------------------------------------------------------------------------

extern "C" void kernel_launch(void* const* d_in, const int* in_sizes, int n_in,
                              void* d_out, int out_size, void* d_ws, size_t ws_size,
                              hipStream_t stream) {
  const float* Whr = (const float*)d_in[0];   // [N, C]
  const float* Whi = (const float*)d_in[1];   // [N, C]
  const int*   Nn  = (const int*)d_in[2];     // [K+1, M]
  // d_in[3] = k_neighbors (device scalar) -- K derived from sizes instead
  const float* wr  = (const float*)d_in[4];   // [C]
  const float* wi  = (const float*)d_in[5];   // [C]
  const float* br  = (const float*)d_in[6];   // [1]
  const float* bi  = (const float*)d_in[7];   // [1]
  float* out = (float*)d_out;                 // [K, M]

  const int C = in_sizes[4];
  const int N = in_sizes[0] / C;
  const int M = in_sizes[2] - out_size;       // (K+1)*M - K*M = M
  const int K = out_size / M;

  float* tmp_r = (float*)d_ws;                // [N]
  float* tmp_i = tmp_r + N;                   // [N]

  {
    dim3 block(256);                           // 8 wave32 -> 128 nodes / block
    dim3 grid((unsigned)((N + 127) / 128));
    gemv_complex_wmma<<<grid, block, 0, stream>>>(Whr, Whi, wr, wi, br, bi,
                                                  tmp_r, tmp_i, N, C);
  }
  {
    dim3 block(256);
    dim3 grid((unsigned)((M + 255) / 256));
    if (K == 32)
      attention_fixed<32><<<grid, block, 0, stream>>>(Nn, tmp_r, tmp_i, out, M);
    else
      attention_generic<<<grid, block, 0, stream>>>(Nn, tmp_r, tmp_i, out, M, K);
  }
}